// GPU_NTM_59184649338919
// MI455X (gfx1250) — compile-verified
//
#include <hip/hip_runtime.h>
#include <math.h>

#define B_   64
#define T_   128
#define NIN  256
#define H_   512
#define O_   256
#define RR   512
#define MM   64
#define EPSF 1e-8f

typedef float v2f __attribute__((ext_vector_type(2)));
typedef float v8f __attribute__((ext_vector_type(8)));

__device__ __forceinline__ float sigmoidf_(float x) { return 1.0f / (1.0f + expf(-x)); }
__device__ __forceinline__ float softplusf_(float x) {
    return fmaxf(x, 0.0f) + log1pf(expf(-fabsf(x)));
}

__device__ __forceinline__ float wave_sum(float v) {
#pragma unroll
    for (int off = 16; off > 0; off >>= 1) v += __shfl_xor(v, off, 32);
    return v;
}
__device__ __forceinline__ float wave_max(float v) {
#pragma unroll
    for (int off = 16; off > 0; off >>= 1) v = fmaxf(v, __shfl_xor(v, off, 32));
    return v;
}
// 256-thread block reductions; deterministic (fixed tree order)
__device__ __forceinline__ float block_sum(float v, float* scr, int tid) {
    v = wave_sum(v);
    __syncthreads();
    if ((tid & 31) == 0) scr[tid >> 5] = v;
    __syncthreads();
    float r = 0.0f;
#pragma unroll
    for (int i = 0; i < 8; ++i) r += scr[i];
    return r;
}
__device__ __forceinline__ float block_max(float v, float* scr, int tid) {
    v = wave_max(v);
    __syncthreads();
    if ((tid & 31) == 0) scr[tid >> 5] = v;
    __syncthreads();
    float r = -3.0e38f;
#pragma unroll
    for (int i = 0; i < 8; ++i) r = fmaxf(r, scr[i]);
    return r;
}

// ---------------------------------------------------------------------------
// Kernel 1: four [64,512]x[512,64] projections via V_WMMA_F32_16X16X4_F32
//   mat 0: k_r = tanh(h Wkr^T + bkr)      mat 1: k_w = tanh(h Wkw^T + bkw)
//   mat 2: e   = sigmoid(h We^T + be)     mat 3: a   = tanh(h Wa^T + ba)
// ---------------------------------------------------------------------------
__global__ __launch_bounds__(32) void proj_kernel(
    const float* __restrict__ h,
    const float* __restrict__ Wkr, const float* __restrict__ bkr,
    const float* __restrict__ Wkw, const float* __restrict__ bkw,
    const float* __restrict__ We,  const float* __restrict__ be,
    const float* __restrict__ Wa,  const float* __restrict__ ba,
    float* __restrict__ kr, float* __restrict__ kw,
    float* __restrict__ ev, float* __restrict__ av)
{
    const int tile = blockIdx.x;          // 0..15 : 4 batch-tiles x 4 col-tiles
    const int mat  = blockIdx.y;          // 0..3
    const int b0   = (tile >> 2) * 16;
    const int j0   = (tile & 3) * 16;
    const int lane = threadIdx.x;
    const int l    = lane & 15;
    const int half = lane >> 4;

    const float* W; const float* bias; float* dst;
    if      (mat == 0) { W = Wkr; bias = bkr; dst = kr; }
    else if (mat == 1) { W = Wkw; bias = bkw; dst = kw; }
    else if (mat == 2) { W = We;  bias = be;  dst = ev; }
    else               { W = Wa;  bias = ba;  dst = av; }

    v8f acc = {};
    const float* arow = h + (size_t)(b0 + l) * H_ + 2 * half;  // A: [m=batch, k]
    const float* brow = W + (size_t)(j0 + l) * H_ + 2 * half;  // B: [k, n=outcol]
#pragma unroll 4
    for (int k0 = 0; k0 < H_; k0 += 4) {
        v2f a2 = *(const v2f*)(arow + k0);
        v2f b2 = *(const v2f*)(brow + k0);
        acc = __builtin_amdgcn_wmma_f32_16x16x4_f32(false, a2, false, b2,
                                                    (short)0, acc, false, false);
    }
    const float bj = bias[j0 + l];
#pragma unroll
    for (int v = 0; v < 8; ++v) {
        int bi = b0 + v + 8 * half;                 // D: vgpr v -> row m
        float x = acc[v] + bj;
        float r = (mat == 2) ? sigmoidf_(x) : tanhf(x);
        dst[bi * MM + (j0 + l)] = r;
    }
}

// ---------------------------------------------------------------------------
// Kernel 2: per-batch addressing + read + memory update (1 WG per batch)
// ---------------------------------------------------------------------------
__global__ __launch_bounds__(256) void mem_kernel(
    float* __restrict__ mem, const float* __restrict__ h,
    float* __restrict__ frf, float* __restrict__ fwf,
    const float* __restrict__ kr, const float* __restrict__ kw,
    const float* __restrict__ ev, const float* __restrict__ av,
    const float* __restrict__ W_br, const float* __restrict__ b_br,
    const float* __restrict__ W_gr, const float* __restrict__ b_gr,
    const float* __restrict__ W_bw, const float* __restrict__ b_bw,
    const float* __restrict__ W_gw, const float* __restrict__ b_gw,
    float* __restrict__ rt)
{
    const int b    = blockIdx.x;
    const int tid  = threadIdx.x;
    const int lane = tid & 31;
    const int wave = tid >> 5;            // 8 waves
    const int m0   = 2 * lane;            // each lane owns 2 M-columns

    __shared__ float s_k[2][MM];
    __shared__ float s_e[MM], s_a[MM];
    __shared__ float s_d[2][RR];          // dots -> scores -> exp
    __shared__ float s_nrm[RR];
    __shared__ float s_f[2][RR];          // focused filters
    __shared__ float s_w[2][RR];          // combined 0.5*(f + wide)
    __shared__ float s_scr[8];
    __shared__ float s_part[8][MM];       // per-wave r_t partials (deterministic)

    if (tid < MM) {
        s_k[0][tid] = kr[b * MM + tid];
        s_k[1][tid] = kw[b * MM + tid];
        s_e[tid]    = ev[b * MM + tid];
        s_a[tid]    = av[b * MM + tid];
    }
    __syncthreads();

    // --- scalar gates beta/g for both heads, plus ||k|| ---
    const float* hb = h + (size_t)b * H_;
    float pbr = 0, pgr = 0, pbw = 0, pgw = 0;
    for (int i = tid; i < H_; i += 256) {
        float hv = hb[i];
        pbr += hv * W_br[i];
        pgr += hv * W_gr[i];
        pbw += hv * W_bw[i];
        pgw += hv * W_gw[i];
    }
    float knr = (tid < MM) ? s_k[0][tid] * s_k[0][tid] : 0.0f;
    float knw = (tid < MM) ? s_k[1][tid] * s_k[1][tid] : 0.0f;

    float beta[2], g[2], knorm[2], s;
    s = block_sum(pbr, s_scr, tid); beta[0]  = softplusf_(s + b_br[0]);
    s = block_sum(pgr, s_scr, tid); g[0]     = sigmoidf_(s + b_gr[0]);
    s = block_sum(pbw, s_scr, tid); beta[1]  = softplusf_(s + b_bw[0]);
    s = block_sum(pgw, s_scr, tid); g[1]     = sigmoidf_(s + b_gw[0]);
    s = block_sum(knr, s_scr, tid); knorm[0] = sqrtf(s);
    s = block_sum(knw, s_scr, tid); knorm[1] = sqrtf(s);

    // --- pass 1: content dots + row norms (one coalesced memory sweep) ---
    float* memb = mem + (size_t)b * RR * MM;
    for (int r = wave; r < RR; r += 8) {
        v2f mv  = *(const v2f*)(memb + (size_t)r * MM + m0);
        float pr = mv.x * s_k[0][m0] + mv.y * s_k[0][m0 + 1];
        float pw = mv.x * s_k[1][m0] + mv.y * s_k[1][m0 + 1];
        float pn = mv.x * mv.x + mv.y * mv.y;
        pr = wave_sum(pr); pw = wave_sum(pw); pn = wave_sum(pn);
        if (lane == 0) { s_d[0][r] = pr; s_d[1][r] = pw; s_nrm[r] = sqrtf(pn); }
    }
    __syncthreads();

    // --- softmax + focused/wide filter update, read and write heads ---
    for (int side = 0; side < 2; ++side) {
        float* fprev = (side == 0 ? frf : fwf) + (size_t)b * RR;
        float lmax = -3.0e38f;
        for (int r = tid; r < RR; r += 256) {
            float sc = beta[side] * s_d[side][r] / (knorm[side] * s_nrm[r] + EPSF);
            s_d[side][r] = sc;
            lmax = fmaxf(lmax, sc);
        }
        __syncthreads();
        float mx = block_max(lmax, s_scr, tid);
        float lsum = 0.0f;
        for (int r = tid; r < RR; r += 256) {
            float e2 = expf(s_d[side][r] - mx);
            s_d[side][r] = e2;
            lsum += e2;
        }
        float inv = 1.0f / block_sum(lsum, s_scr, tid);
        for (int r = tid; r < RR; r += 256) {
            float wc = s_d[side][r] * inv;
            float fn = g[side] * wc + (1.0f - g[side]) * fprev[r];
            s_f[side][r] = fn;
            fprev[r] = fn;                      // persist focused-filter state
        }
        __syncthreads();
        for (int r = tid; r < RR; r += 256) {
            float wide = 0.25f * s_f[side][(r + RR - 1) & (RR - 1)]
                       + 0.50f * s_f[side][r]
                       + 0.25f * s_f[side][(r + 1) & (RR - 1)];
            s_w[side][r] = 0.5f * (s_f[side][r] + wide);
        }
        __syncthreads();
    }

    // --- pass 2: fused read r_t + erase/add update (one sweep, in place) ---
    float acc0 = 0.0f, acc1 = 0.0f;
    const float e0 = s_e[m0], e1 = s_e[m0 + 1];
    const float a0 = s_a[m0], a1 = s_a[m0 + 1];
    for (int r = wave; r < RR; r += 8) {
        float* rowp = memb + (size_t)r * MM + m0;
        v2f mv = *(const v2f*)rowp;
        float wr = s_w[0][r];
        float ww = s_w[1][r];
        acc0 += wr * mv.x;
        acc1 += wr * mv.y;
        v2f nv;
        nv.x = mv.x * (1.0f - ww * e0) + ww * a0;
        nv.y = mv.y * (1.0f - ww * e1) + ww * a1;
        *(v2f*)rowp = nv;
    }
    s_part[wave][m0]     = acc0;
    s_part[wave][m0 + 1] = acc1;
    __syncthreads();
    if (tid < MM) {
        float t = 0.0f;
#pragma unroll
        for (int wv = 0; wv < 8; ++wv) t += s_part[wv][tid];
        rt[b * MM + tid] = t;
    }
}

// ---------------------------------------------------------------------------
// Kernel 3: h = relu(x_t W_in^T + r_t W_read^T + b_in + b_read)  (WMMA f32)
// ---------------------------------------------------------------------------
__global__ __launch_bounds__(32) void ctrl_kernel(
    const float* __restrict__ x, const float* __restrict__ rt,
    const float* __restrict__ W_in, const float* __restrict__ b_in,
    const float* __restrict__ W_read, const float* __restrict__ b_read,
    float* __restrict__ h, int t)
{
    const int b0   = blockIdx.x * 16;     // 0..3
    const int j0   = blockIdx.y * 16;     // 0..31
    const int lane = threadIdx.x;
    const int l    = lane & 15;
    const int half = lane >> 4;

    v8f acc = {};
    const float* arow = x + (size_t)(b0 + l) * (T_ * NIN) + (size_t)t * NIN + 2 * half;
    const float* brow = W_in + (size_t)(j0 + l) * NIN + 2 * half;
#pragma unroll 4
    for (int k0 = 0; k0 < NIN; k0 += 4) {
        v2f a2 = *(const v2f*)(arow + k0);
        v2f b2 = *(const v2f*)(brow + k0);
        acc = __builtin_amdgcn_wmma_f32_16x16x4_f32(false, a2, false, b2,
                                                    (short)0, acc, false, false);
    }
    const float* arow2 = rt + (size_t)(b0 + l) * MM + 2 * half;
    const float* brow2 = W_read + (size_t)(j0 + l) * MM + 2 * half;
#pragma unroll 4
    for (int k0 = 0; k0 < MM; k0 += 4) {
        v2f a2 = *(const v2f*)(arow2 + k0);
        v2f b2 = *(const v2f*)(brow2 + k0);
        acc = __builtin_amdgcn_wmma_f32_16x16x4_f32(false, a2, false, b2,
                                                    (short)0, acc, false, false);
    }
    const float bj = b_in[j0 + l] + b_read[j0 + l];
#pragma unroll
    for (int v = 0; v < 8; ++v) {
        int bi = b0 + v + 8 * half;
        float xx = acc[v] + bj;
        h[(size_t)bi * H_ + (j0 + l)] = xx > 0.0f ? xx : 0.0f;
    }
}

// ---------------------------------------------------------------------------
// Kernel 4: out[:,t,:] = sigmoid(h W_out^T + b_out)  (WMMA f32)
// ---------------------------------------------------------------------------
__global__ __launch_bounds__(32) void out_kernel(
    const float* __restrict__ h,
    const float* __restrict__ W_out, const float* __restrict__ b_out,
    float* __restrict__ out, int t)
{
    const int b0   = blockIdx.x * 16;     // 0..3
    const int o0   = blockIdx.y * 16;     // 0..15
    const int lane = threadIdx.x;
    const int l    = lane & 15;
    const int half = lane >> 4;

    v8f acc = {};
    const float* arow = h + (size_t)(b0 + l) * H_ + 2 * half;
    const float* brow = W_out + (size_t)(o0 + l) * H_ + 2 * half;
#pragma unroll 4
    for (int k0 = 0; k0 < H_; k0 += 4) {
        v2f a2 = *(const v2f*)(arow + k0);
        v2f b2 = *(const v2f*)(brow + k0);
        acc = __builtin_amdgcn_wmma_f32_16x16x4_f32(false, a2, false, b2,
                                                    (short)0, acc, false, false);
    }
    const float bo = b_out[o0 + l];
#pragma unroll
    for (int v = 0; v < 8; ++v) {
        int bi = b0 + v + 8 * half;
        out[(size_t)bi * (T_ * O_) + (size_t)t * O_ + (o0 + l)] =
            sigmoidf_(acc[v] + bo);
    }
}

// ---------------------------------------------------------------------------
extern "C" void kernel_launch(void* const* d_in, const int* in_sizes, int n_in,
                              void* d_out, int out_size, void* d_ws, size_t ws_size,
                              hipStream_t stream)
{
    (void)in_sizes; (void)n_in; (void)out_size; (void)ws_size;
    const float* x      = (const float*)d_in[0];
    const float* h0     = (const float*)d_in[1];
    const float* frf0   = (const float*)d_in[2];
    // d_in[3] = wrf, d_in[5] = wwf: dead state (wide filters are recomputed)
    const float* fwf0   = (const float*)d_in[4];
    const float* mem0   = (const float*)d_in[6];
    const float* W_in   = (const float*)d_in[7];
    const float* b_in   = (const float*)d_in[8];
    const float* W_read = (const float*)d_in[9];
    const float* b_read = (const float*)d_in[10];
    const float* W_out  = (const float*)d_in[11];
    const float* b_out  = (const float*)d_in[12];
    const float* W_kr   = (const float*)d_in[13];
    const float* b_kr   = (const float*)d_in[14];
    const float* W_br   = (const float*)d_in[15];
    const float* b_br   = (const float*)d_in[16];
    const float* W_gr   = (const float*)d_in[17];
    const float* b_gr   = (const float*)d_in[18];
    const float* W_kw   = (const float*)d_in[19];
    const float* b_kw   = (const float*)d_in[20];
    const float* W_bw   = (const float*)d_in[21];
    const float* b_bw   = (const float*)d_in[22];
    const float* W_gw   = (const float*)d_in[23];
    const float* b_gw   = (const float*)d_in[24];
    const float* W_e    = (const float*)d_in[25];
    const float* b_e    = (const float*)d_in[26];
    const float* W_a    = (const float*)d_in[27];
    const float* b_a    = (const float*)d_in[28];
    float* out = (float*)d_out;

    // workspace layout (floats); state lives here, inputs stay untouched
    float* w   = (float*)d_ws;
    float* mem = w;                    // 64*512*64 = 2,097,152
    float* h   = mem + 2097152;        // 64*512    =    32,768
    float* frf = h   + 32768;          // 64*512
    float* fwf = frf + 32768;          // 64*512
    float* kr  = fwf + 32768;          // 64*64
    float* kw  = kr  + 4096;
    float* ev  = kw  + 4096;
    float* av  = ev  + 4096;
    float* rt  = av  + 4096;           // 64*64

    hipMemcpyAsync(mem, mem0, (size_t)2097152 * sizeof(float), hipMemcpyDeviceToDevice, stream);
    hipMemcpyAsync(h,   h0,   (size_t)32768   * sizeof(float), hipMemcpyDeviceToDevice, stream);
    hipMemcpyAsync(frf, frf0, (size_t)32768   * sizeof(float), hipMemcpyDeviceToDevice, stream);
    hipMemcpyAsync(fwf, fwf0, (size_t)32768   * sizeof(float), hipMemcpyDeviceToDevice, stream);

    for (int t = 0; t < T_; ++t) {
        proj_kernel<<<dim3(16, 4), 32, 0, stream>>>(h,
            W_kr, b_kr, W_kw, b_kw, W_e, b_e, W_a, b_a, kr, kw, ev, av);
        mem_kernel<<<B_, 256, 0, stream>>>(mem, h, frf, fwf, kr, kw, ev, av,
            W_br, b_br, W_gr, b_gr, W_bw, b_bw, W_gw, b_gw, rt);
        ctrl_kernel<<<dim3(4, 32), 32, 0, stream>>>(x, rt, W_in, b_in,
            W_read, b_read, h, t);
        out_kernel<<<dim3(4, 16), 32, 0, stream>>>(h, W_out, b_out, out, t);
    }
}